// KGERuleFilter_66460323938770
// MI455X (gfx1250) — compile-verified
//
#include <hip/hip_runtime.h>
#include <stdint.h>

// KGE rule scoring + per-row top-64 for MI455X (gfx1250, wave32).
// Scores computed as diagonals of V_WMMA_F32_16X16X4_F32 accumulations.
// Inner loop uses b128 gathers: each lane loads float4 at d + (upperhalf?4:0)
// and feeds .lo/.hi to two WMMA steps (valid d->K-slot permutation since both
// A and B fragments source K-slots {0,1} from lanes 0-15 and {2,3} from 16-31).

typedef __attribute__((ext_vector_type(2))) float v2f;
typedef __attribute__((ext_vector_type(4))) float v4f;
typedef __attribute__((ext_vector_type(8))) float v8f;

#define B_DIM 4
#define S_DIM 32
#define KR 1024
#define D_DIM 512
#define TOPK 64
#define CONST_NO 49999
#define NEG_INF_F (-1000000000.0f)

__device__ __forceinline__ unsigned int orderKey(float f) {
  // monotone map: larger float -> larger uint
  unsigned int u = __float_as_uint(f);
  return (u & 0x80000000u) ? ~u : (u | 0x80000000u);
}

__global__ __launch_bounds__(256) void kge_score_topk(
    const int* __restrict__ rule_goals,             // (B,S,KR,2,3) int32
    const unsigned char* __restrict__ rule_success, // (B,S,KR) bool
    const float* __restrict__ ent_emb,              // (60000,512)
    const float* __restrict__ rel_emb,              // (512,512)
    float* __restrict__ out_mask,                   // (B,S,KR) as 0/1 float
    float* __restrict__ out_scores)                 // (B,S,KR) float
{
  __shared__ unsigned int s_keys[KR];
  __shared__ unsigned char s_succ[KR];
  __shared__ unsigned int s_cnt;

  const int row  = blockIdx.x;        // 0..127 = b*S+s
  const int tid  = threadIdx.x;       // 0..255
  const int wave = tid >> 5;          // 0..7
  const int lane = tid & 31;
  const int mrule = lane & 15;        // rule-in-group serviced by this lane
  const bool lower = (lane < 16);
  const int off4 = lower ? 0 : 4;     // float4 base offset for this half-wave

  const long long goalRowBase = (long long)row * KR * 6; // atom0 stride = 6 ints
  const long long scoreRowBase = (long long)row * KR;

  // ---------------- Phase A: scores via WMMA (8 groups of 16 rules / wave) ---
  for (int g = 0; g < 8; ++g) {
    const int k = wave * 128 + g * 16 + mrule;
    const long long gb = goalRowBase + (long long)k * 6;
    const int p  = rule_goals[gb + 0];
    const int a1 = rule_goals[gb + 1];
    const int a2 = rule_goals[gb + 2];
    const bool ground = (a1 <= CONST_NO) && (a2 <= CONST_NO) && (p != 0);
    const int sp  = ground ? p  : 0;
    const int sa1 = ground ? a1 : 0;
    const int sa2 = ground ? a2 : 0;

    const float* __restrict__ pa1 = ent_emb + (size_t)sa1 * D_DIM + off4;
    const float* __restrict__ pp  = rel_emb + (size_t)sp  * D_DIM + off4;
    const float* __restrict__ pa2 = ent_emb + (size_t)sa2 * D_DIM + off4;

    // C(16x16) += A(16x4) * B(4x16); diagonal = per-rule triple dot products.
    v8f c = {};
#pragma unroll 4
    for (int d = 0; d < D_DIM; d += 8) {
      v4f ea1 = *(const v4f*)(pa1 + d);   // global_load_b128
      v4f ep  = *(const v4f*)(pp  + d);
      v4f ea2 = *(const v4f*)(pa2 + d);
      v4f u = ea1 * ep;                   // U[m,d] = ent[a1]*rel[p]
      // step 0: K-slots <- lower {d,d+1}, upper {d+4,d+5}
      c = __builtin_amdgcn_wmma_f32_16x16x4_f32(
          false, u.lo, false, ea2.lo, (short)0, c, false, false);
      // step 1: K-slots <- lower {d+2,d+3}, upper {d+6,d+7}
      c = __builtin_amdgcn_wmma_f32_16x16x4_f32(
          false, u.hi, false, ea2.hi, (short)0, c, false, false);
    }

    // Diagonal extraction: row m lives in VGPR m (lanes 0-15) / VGPR m-8 (16-31).
    float sval = 0.0f;
    bool diag = false;
#pragma unroll
    for (int r = 0; r < 8; ++r) {
      const int m = lower ? r : (r + 8);
      const int n = lower ? lane : (lane - 16);
      if (m == n) { sval = c[r]; diag = true; }
    }
    if (diag) { // lanes 0-7 own rules 0-7; lanes 24-31 own rules 8-15
      const unsigned char su = rule_success[scoreRowBase + k];
      float s = ground ? sval : 0.0f;
      s = su ? s : NEG_INF_F;
      out_scores[scoreRowBase + k] = s;
      s_keys[k] = orderKey(s);
      s_succ[k] = su;
    }
  }
  __syncthreads();

  // ---------- Phase B: binary search over key bits for 64th-largest ----------
  unsigned int lo = 0u;
  for (int b = 31; b >= 0; --b) {
    const unsigned int mid = lo | (1u << b);
    if (tid == 0) s_cnt = 0u;
    __syncthreads();
    unsigned int local = 0u;
    for (int i = tid; i < KR; i += 256) local += (s_keys[i] >= mid) ? 1u : 0u;
    atomicAdd(&s_cnt, local);
    __syncthreads();
    if (s_cnt >= TOPK) lo = mid;   // keep count(key >= lo) >= 64
    __syncthreads();
  }
  const unsigned int T = lo;       // 64th-largest key

  if (tid == 0) s_cnt = 0u;
  __syncthreads();
  {
    unsigned int local = 0u;
    for (int i = tid; i < KR; i += 256) local += (s_keys[i] > T) ? 1u : 0u;
    atomicAdd(&s_cnt, local);
  }
  __syncthreads();
  const int quota = TOPK - (int)s_cnt; // slots for keys == T, lowest-index first

  // ---------------- Phase C: emit mask = rule_success & top-64 --------------
  for (int i = tid; i < KR; i += 256) {
    const unsigned int key = s_keys[i];
    bool keep = false;
    if (key > T) {
      keep = true;
    } else if (key == T) {
      int rank = 0;
      for (int j = 0; j < i; ++j) rank += (s_keys[j] == T) ? 1 : 0;
      keep = (rank < quota);       // deterministic lowest-index tie break
    }
    const bool m = keep && (s_succ[i] != 0);
    out_mask[scoreRowBase + i] = m ? 1.0f : 0.0f;
  }
}

extern "C" void kernel_launch(void* const* d_in, const int* in_sizes, int n_in,
                              void* d_out, int out_size, void* d_ws, size_t ws_size,
                              hipStream_t stream) {
  (void)in_sizes; (void)n_in; (void)d_ws; (void)ws_size; (void)out_size;
  const int*           rule_goals   = (const int*)d_in[0];
  const unsigned char* rule_success = (const unsigned char*)d_in[1];
  // d_in[2] = queries (unused by reference computation)
  const float*         ent_emb      = (const float*)d_in[3];
  const float*         rel_emb      = (const float*)d_in[4];

  float* out_mask   = (float*)d_out;                      // first output (bool as 0/1)
  float* out_scores = out_mask + (B_DIM * S_DIM * KR);    // second output

  kge_score_topk<<<B_DIM * S_DIM, 256, 0, stream>>>(
      rule_goals, rule_success, ent_emb, rel_emb, out_mask, out_scores);
}